// TorsoGCNv1_78168404787865
// MI455X (gfx1250) — compile-verified
//
#include <hip/hip_runtime.h>
#include <hip/hip_bf16.h>

typedef __attribute__((ext_vector_type(2))) float v2f;
typedef __attribute__((ext_vector_type(8))) float v8f;

#define NNODES 40000
#define NEDGES 640000
#define NGRAPH 256

__device__ __forceinline__ void atomAddF(float* p, float v) {
    unsafeAtomicAdd(p, v);   // lowers to global_atomic_add_f32 (no CAS loop)
}

// ---- degree: deg[dst[e]] += 1 ------------------------------------------------
__global__ void deg_kernel(const int* __restrict__ dst, float* __restrict__ deg, int E) {
    int e = blockIdx.x * blockDim.x + threadIdx.x;
    if (e < E) atomAddF(&deg[dst[e]], 1.0f);
}

// ---- dis = rsqrt(deg+1); counts[batch[n]] += 1 ------------------------------
__global__ void dis_counts_kernel(float* __restrict__ deg_dis,
                                  const int* __restrict__ batch,
                                  float* __restrict__ counts, int N) {
    int n = blockIdx.x * blockDim.x + threadIdx.x;
    if (n < N) {
        deg_dis[n] = rsqrtf(deg_dis[n] + 1.0f);
        atomAddF(&counts[batch[n]], 1.0f);
    }
}

// ---- fp32 WMMA GEMM: C[N x M] = A[N x K] @ B[K x M], optional row-scale/bias
// one wave per 16x16 output tile; V_WMMA_F32_16X16X4_F32 over K in steps of 4.
__global__ void wmma_gemm_f32(const float* __restrict__ A, const float* __restrict__ B,
                              float* __restrict__ C, int N, int K, int M,
                              const float* __restrict__ rowscale,
                              const float* __restrict__ bias, int total_tiles) {
    int tid  = threadIdx.x;
    int wid  = tid >> 5;
    int lane = tid & 31;
    int tile = blockIdx.x * (blockDim.x >> 5) + wid;
    if (tile >= total_tiles) return;               // wave-uniform branch

    int tiles_m = M >> 4;
    int rt = tile / tiles_m, ct = tile - rt * tiles_m;
    int row0 = rt << 4, col0 = ct << 4;
    int hi = lane >> 4, lm = lane & 15;

    v8f c = {};
    const float* Arow = A + (size_t)(row0 + lm) * K;   // A: M=lm, K = v + 2*hi
    for (int k0 = 0; k0 < K; k0 += 4) {
        v2f a = *reinterpret_cast<const v2f*>(Arow + k0 + 2 * hi);   // 8B aligned
        const float* Bp = B + (size_t)(k0 + 2 * hi) * M + col0 + lm; // B: K = v + 2*hi, N=lm
        v2f b;
        b.x = Bp[0];
        b.y = Bp[M];
        c = __builtin_amdgcn_wmma_f32_16x16x4_f32(false, a, false, b,
                                                  (short)0, c, false, false);
    }
    float bv = bias ? bias[col0 + lm] : 0.0f;
    #pragma unroll
    for (int v = 0; v < 8; ++v) {                  // C/D: row = row0 + v + 8*hi
        int row = row0 + v + 8 * hi;
        float o = c[v];
        if (rowscale) o *= rowscale[row];
        C[(size_t)row * M + col0 + lm] = o + bv;
    }
}

// ---- edge scatter: agg[dst[e]][:] += h[src[e]][:], one wave per edge --------
__global__ void scatter_kernel(const float* __restrict__ h, const int* __restrict__ src,
                               const int* __restrict__ dst, float* __restrict__ agg,
                               int F, int E) {
    int tid  = blockIdx.x * blockDim.x + threadIdx.x;
    int e    = tid >> 5;
    int lane = tid & 31;
    if (e >= E) return;
    int s = src[e], d = dst[e];
    const float* hs = h + (size_t)s * F;
    float* ad = agg + (size_t)d * F;
    for (int f = lane; f < F; f += 32)
        atomAddF(&ad[f], hs[f]);
}

// ---- agg = relu((agg + h) * dis[n] + b[f]), in place ------------------------
__global__ void combine_kernel(float* __restrict__ agg, const float* __restrict__ h,
                               const float* __restrict__ dis, const float* __restrict__ bias,
                               int fshift, int total) {
    int i = blockIdx.x * blockDim.x + threadIdx.x;
    if (i >= total) return;
    int n = i >> fshift;
    int f = i & ((1 << fshift) - 1);
    float v = (agg[i] + h[i]) * dis[n] + bias[f];
    agg[i] = v > 0.0f ? v : 0.0f;
}

// ---- mean-pool accumulate: pooled[batch[n]][f] += h[n][f]  (F == 32) --------
__global__ void pool_kernel(const float* __restrict__ h, const int* __restrict__ batch,
                            float* __restrict__ pooled, int total) {
    int i = blockIdx.x * blockDim.x + threadIdx.x;
    if (i >= total) return;
    int n = i >> 5, f = i & 31;
    atomAddF(&pooled[(batch[n] << 5) + f], h[i]);
}

// ---- pooled /= max(counts, 1) ----------------------------------------------
__global__ void poolfin_kernel(float* __restrict__ pooled, const float* __restrict__ counts) {
    int i = blockIdx.x * blockDim.x + threadIdx.x;   // 256*32 = 8192 threads
    int g = i >> 5;
    pooled[i] = pooled[i] / fmaxf(counts[g], 1.0f);
}

extern "C" void kernel_launch(void* const* d_in, const int* in_sizes, int n_in,
                              void* d_out, int out_size, void* d_ws, size_t ws_size,
                              hipStream_t stream) {
    // setup_inputs order: x, W1, b1, W2, b2, W3, b3, Wl, bl, edge_index, batch
    const float* x   = (const float*)d_in[0];
    const float* W1  = (const float*)d_in[1];
    const float* b1  = (const float*)d_in[2];
    const float* W2  = (const float*)d_in[3];
    const float* b2  = (const float*)d_in[4];
    const float* W3  = (const float*)d_in[5];
    const float* b3  = (const float*)d_in[6];
    const float* Wl  = (const float*)d_in[7];
    const float* bl  = (const float*)d_in[8];
    const int*   ei  = (const int*)d_in[9];     // [2, E] flat: row0 = src, row1 = dst
    const int*   bat = (const int*)d_in[10];
    float*       out = (float*)d_out;           // [256, 768] fp32

    const int N = NNODES, E = NEDGES;
    const int* src = ei;
    const int* dst = ei + E;

    // workspace layout (fp32): two rotating N x 128 buffers + small arrays (~41.2 MB)
    float* B0     = (float*)d_ws;               // N*128
    float* B1     = B0 + (size_t)N * 128;       // N*128
    float* dis    = B1 + (size_t)N * 128;       // N   (deg, then rsqrt in place)
    float* counts = dis + N;                    // 256
    float* pooled = counts + NGRAPH;            // 256*32

    // degree + norm + graph counts
    hipMemsetAsync(dis, 0, (size_t)N * sizeof(float), stream);
    hipMemsetAsync(counts, 0, NGRAPH * sizeof(float), stream);
    deg_kernel<<<E / 256, 256, 0, stream>>>(dst, dis, E);
    dis_counts_kernel<<<(N + 255) / 256, 256, 0, stream>>>(dis, bat, counts, N);

    // ---- layer 1: 128 -> 128 ----
    wmma_gemm_f32<<<(2500 * 8) / 8, 256, 0, stream>>>(x, W1, B0, N, 128, 128, dis, nullptr, 2500 * 8);
    hipMemsetAsync(B1, 0, (size_t)N * 128 * sizeof(float), stream);
    scatter_kernel<<<(E * 32) / 256, 256, 0, stream>>>(B0, src, dst, B1, 128, E);
    combine_kernel<<<(N * 128) / 256, 256, 0, stream>>>(B1, B0, dis, b1, 7, N * 128);

    // ---- layer 2: 128 -> 64 ----
    wmma_gemm_f32<<<(2500 * 4) / 8, 256, 0, stream>>>(B1, W2, B0, N, 128, 64, dis, nullptr, 2500 * 4);
    hipMemsetAsync(B1, 0, (size_t)N * 64 * sizeof(float), stream);
    scatter_kernel<<<(E * 32) / 256, 256, 0, stream>>>(B0, src, dst, B1, 64, E);
    combine_kernel<<<(N * 64) / 256, 256, 0, stream>>>(B1, B0, dis, b2, 6, N * 64);

    // ---- layer 3: 64 -> 32 ----
    wmma_gemm_f32<<<(2500 * 2) / 8, 256, 0, stream>>>(B1, W3, B0, N, 64, 32, dis, nullptr, 2500 * 2);
    hipMemsetAsync(B1, 0, (size_t)N * 32 * sizeof(float), stream);
    scatter_kernel<<<(E * 32) / 256, 256, 0, stream>>>(B0, src, dst, B1, 32, E);
    combine_kernel<<<(N * 32) / 256, 256, 0, stream>>>(B1, B0, dis, b3, 5, N * 32);

    // ---- global mean pool ----
    hipMemsetAsync(pooled, 0, NGRAPH * 32 * sizeof(float), stream);
    pool_kernel<<<(N * 32) / 256, 256, 0, stream>>>(B1, bat, pooled, N * 32);
    poolfin_kernel<<<(NGRAPH * 32) / 256, 256, 0, stream>>>(pooled, counts);

    // ---- head: [256,32] @ [32,768] + bl -> d_out ----
    wmma_gemm_f32<<<(16 * 48) / 8, 256, 0, stream>>>(pooled, Wl, out, NGRAPH, 32, 768,
                                                     nullptr, bl, 16 * 48);
}